// SinkhornLayer_36051955482723
// MI455X (gfx1250) — compile-verified
//
#include <hip/hip_runtime.h>
#include <math.h>

// Sinkhorn layer for MI455X (gfx1250).
// A_t = A0 - R[i] - C[j]  (rank-1 log-domain rewrite) => A0 is read-only and
// L2-resident (134MB < 192MB L2). 128 row-major LSE passes (A0 symmetric, so
// both axes reduce along rows); WMMA f32 16x16x4 with ones-A accumulates the
// row sums of exp(); vector staged to LDS via async-to-LDS when available.

#define NB   32
#define LL   1024
#define DD   1023
#define PAD_NEG (-1e30f)
#define PAD_POS (1e30f)

typedef __attribute__((ext_vector_type(2))) float v2f;
typedef __attribute__((ext_vector_type(8))) float v8f;
typedef __attribute__((__vector_size__(4 * sizeof(int)))) int v4i_t;

#define AS1 __attribute__((address_space(1)))
#define AS3 __attribute__((address_space(3)))

#if defined(__has_builtin)
#if __has_builtin(__builtin_amdgcn_global_load_async_to_lds_b128)
#define HAVE_ASYNC_B128 1
#endif
#if __has_builtin(__builtin_amdgcn_global_load_async_to_lds_b32)
#define HAVE_ASYNC_B32 1
#endif
#if __has_builtin(__builtin_amdgcn_s_wait_asynccnt)
#define WAIT_ASYNC() __builtin_amdgcn_s_wait_asynccnt(0)
#else
#define WAIT_ASYNC() asm volatile("s_wait_asynccnt 0x0" ::: "memory")
#endif
#endif

__device__ __forceinline__ float gumbel(float n) {
  float u = 1e-5f + n * (1.0f - 1e-5f);
  return -__logf(-__logf(u));
}

// ---------------------------------------------------------------- build A0
// A0[b,i,j] (padded to 1024x1024, pad = -1e30):
//   i==j : min(xu[b,1+i],50) + g(noise[i,i])
//   i<j  : min(xp[b,1+i,1+j],50) + 0.5*g(noise[i,j])   (mirrored for i>j)
__global__ void build_A0(const float* __restrict__ xp,
                         const float* __restrict__ xu,
                         const float* __restrict__ noise,
                         float* __restrict__ A0) {
  __shared__ float tile[64][65];
  int bid = blockIdx.x;
  int b = bid / 136;
  int t = bid % 136;
  int ti = 0;
  while (t >= 16 - ti) { t -= 16 - ti; ++ti; }
  int tj = ti + t;

  int tid = threadIdx.x;
  int tx = tid & 63;          // column within tile
  int tyBase = tid >> 6;      // 0..3

  const float* xpB = xp + (size_t)b * LL * LL;
  const float* nsB = noise + (size_t)b * DD * DD;
  const float* xuB = xu + b * LL;
  float* A0B = A0 + (size_t)b * LL * LL;

  int jj = tj * 64 + tx;
  for (int q = 0; q < 16; ++q) {
    int r = tyBase * 16 + q;
    int ii = ti * 64 + r;
    int lo = min(ii, jj), hi = max(ii, jj);
    float val;
    if (hi >= DD) {
      val = PAD_NEG;
    } else {
      float g = gumbel(nsB[lo * DD + hi]);
      if (lo == hi) val = fminf(xuB[1 + lo], 50.0f) + g;
      else          val = fminf(xpB[(1 + lo) * LL + (1 + hi)], 50.0f) + 0.5f * g;
    }
    A0B[(size_t)ii * LL + jj] = val;
    tile[r][tx] = val;
  }
  __syncthreads();
  if (ti != tj) {
    for (int q = 0; q < 16; ++q) {
      int r = tyBase * 16 + q;
      int ii2 = tj * 64 + r;
      int jj2 = ti * 64 + tx;
      A0B[(size_t)ii2 * LL + jj2] = tile[tx][r];   // bank-conflict-free (65 pad)
    }
  }
}

// ---------------------------------------------------------------- init R=0
__global__ void init_vec(float* __restrict__ v) {
  int i = blockIdx.x * 256 + threadIdx.x;           // 32*1024
  int k = i & (LL - 1);
  v[i] = (k == DD) ? PAD_POS : 0.0f;                // pad col pinned high
}

// ---------------------------------------------------------------- LSE phase
// vout[b,row] = LSE_k( A0[b,row,k] - vin[b,k] ).
// Wave owns 16 rows. Lane layout matches WMMA f32 16x16x4 B-operand:
//   lane n / n+16 -> row n; VGPR pair -> cols (4c + 2*half), (4c + 2*half + 1).
// Pass 1: row max (shfl_xor 16 merges lane halves).
// Pass 2: e = exp(a - v - m); dual WMMA accumulators (ones 16x4 A-matrix)
//         accumulate all 16 row sums; acc[0] in lane n holds row n's sum.
__global__ void lse_phase(const float* __restrict__ A0,
                          const float* __restrict__ vin,
                          float* __restrict__ vout) {
  __shared__ __align__(16) float vec[LL];
  int b = blockIdx.x >> 3;                // 8 blocks per batch
  int blkRow = (blockIdx.x & 7) * 128;    // 8 waves * 16 rows
  int tid = threadIdx.x;

  const float* vinB = vin + b * LL;
#if defined(HAVE_ASYNC_B128)
  // 256 threads x 16B = 4KB, single async DMA each; tracked by ASYNCcnt.
  __builtin_amdgcn_global_load_async_to_lds_b128(
      (v4i_t*)(vinB + 4 * tid), (AS3 v4i_t*)(vec + 4 * tid), 0, 0);
  WAIT_ASYNC();
#elif defined(HAVE_ASYNC_B32)
  for (int i = tid; i < LL; i += 256)
    __builtin_amdgcn_global_load_async_to_lds_b32(
        (int*)(vinB + i), (AS3 int*)(vec + i), 0, 0);
  WAIT_ASYNC();
#else
  for (int i = tid; i < LL; i += 256) vec[i] = vinB[i];
#endif
  __syncthreads();

  int wave = tid >> 5;
  int lane = tid & 31;
  int half = lane >> 4;                   // 0/1 -> K offset 0/2
  int mrow = lane & 15;
  int rowBase = blkRow + wave * 16;
  const float* rowPtr =
      A0 + (size_t)b * LL * LL + (size_t)(rowBase + mrow) * LL + half * 2;
  int voff = half * 2;

  // pass 1: row max of (a - v)
  float mx = -INFINITY;
  #pragma unroll 8
  for (int c = 0; c < 256; ++c) {
    v2f a = *(const v2f*)(rowPtr + 4 * c);
    float vx = vec[4 * c + voff];
    float vy = vec[4 * c + voff + 1];
    mx = fmaxf(mx, fmaxf(a.x - vx, a.y - vy));
  }
  mx = fmaxf(mx, __shfl_xor(mx, 16, 32));  // merge the two lane halves

  // pass 2: sum of exp via WMMA (A = ones 16x4, B = exp data 4x16)
  v2f ones; ones.x = 1.0f; ones.y = 1.0f;
  v8f acc0 = {};
  v8f acc1 = {};
  #pragma unroll 2
  for (int c = 0; c < 256; c += 2) {
    v2f a0 = *(const v2f*)(rowPtr + 4 * c);
    v2f a1 = *(const v2f*)(rowPtr + 4 * c + 4);
    float v0x = vec[4 * c + voff];
    float v0y = vec[4 * c + voff + 1];
    float v1x = vec[4 * c + 4 + voff];
    float v1y = vec[4 * c + 4 + voff + 1];
    v2f e0, e1;
    e0.x = __expf((a0.x - v0x) - mx);
    e0.y = __expf((a0.y - v0y) - mx);
    e1.x = __expf((a1.x - v1x) - mx);
    e1.y = __expf((a1.y - v1y) - mx);
    acc0 = __builtin_amdgcn_wmma_f32_16x16x4_f32(
        false, ones, false, e0, (short)0, acc0, false, false);
    acc1 = __builtin_amdgcn_wmma_f32_16x16x4_f32(
        false, ones, false, e1, (short)0, acc1, false, false);
  }
  float s = acc0[0] + acc1[0];             // D[*, N=mrow] = sum of row mrow

  if (lane < 16) {
    int row = rowBase + mrow;
    float lse = mx + __logf(s);
    vout[b * LL + row] = (row == DD) ? PAD_POS : lse;  // keep pad pinned high
  }
}

// ---------------------------------------------------------------- finalize
__global__ void finalize_xp(const float* __restrict__ xpin,
                            const float* __restrict__ A0,
                            const float* __restrict__ Rv,
                            const float* __restrict__ Cv,
                            float* __restrict__ out) {
  size_t idx = (size_t)blockIdx.x * 256 + threadIdx.x;
  int b = (int)(idx >> 20);
  int rem = (int)(idx & ((1u << 20) - 1));
  int i = rem >> 10;
  int j = rem & 1023;
  float o;
  if (i == 0 || j == 0) {
    float v = fminf(xpin[idx], 50.0f);
    o = (i == 0 && j == 0) ? __expf(v) : v;
  } else {
    int ii = i - 1, jj = j - 1;
    if (ii == jj) {
      o = 0.0f;  // triu(k=1)+transpose has zero diagonal
    } else {
      o = __expf(A0[((size_t)b << 20) + ((size_t)ii << 10) + jj]
                 - Rv[(b << 10) + ii] - Cv[(b << 10) + jj]);
    }
  }
  out[idx] = o;
}

__global__ void finalize_xu(const float* __restrict__ xuin,
                            const float* __restrict__ A0,
                            const float* __restrict__ Rv,
                            const float* __restrict__ Cv,
                            float* __restrict__ out) {
  int idx = blockIdx.x * 256 + threadIdx.x;          // 32*1024
  int b = idx >> 10, k = idx & 1023;
  float o;
  if (k == 0) {
    o = __expf(fminf(xuin[idx], 50.0f));
  } else {
    int kk = k - 1;
    o = __expf(A0[((size_t)b << 20) + ((size_t)kk << 10) + kk]
               - Rv[(b << 10) + kk] - Cv[(b << 10) + kk]);
  }
  out[(size_t)NB * LL * LL + idx] = o;
}

// ---------------------------------------------------------------- launch
extern "C" void kernel_launch(void* const* d_in, const int* in_sizes, int n_in,
                              void* d_out, int out_size, void* d_ws, size_t ws_size,
                              hipStream_t stream) {
  const float* xp = (const float*)d_in[0];     // (32,1024,1024)
  const float* xu = (const float*)d_in[1];     // (32,1024)
  const float* noise = (const float*)d_in[2];  // (32,1023,1023)

  float* A0 = (float*)d_ws;                                // 134.2 MB
  float* vecR = A0 + (size_t)NB * LL * LL;                 // 128 KB
  float* vecC = vecR + NB * LL;                            // 128 KB
  float* out = (float*)d_out;

  build_A0<<<NB * 136, 256, 0, stream>>>(xp, xu, noise, A0);
  init_vec<<<(NB * LL) / 256, 256, 0, stream>>>(vecR);

  for (int t = 0; t < 64; ++t) {
    // axis=1 first: C[j] = LSE_i(A0 - R)  (row-major via symmetry of A0)
    lse_phase<<<NB * 8, 256, 0, stream>>>(A0, vecR, vecC);
    // axis=2: R[i] = LSE_j(A0 - C)
    lse_phase<<<NB * 8, 256, 0, stream>>>(A0, vecC, vecR);
  }

  finalize_xp<<<(NB * LL * LL) / 256, 256, 0, stream>>>(xp, A0, vecR, vecC, out);
  finalize_xu<<<(NB * LL) / 256, 256, 0, stream>>>(xu, A0, vecR, vecC, out);
}